// DFineMultiScaleDeformableAttention_11235634446733
// MI455X (gfx1250) — compile-verified
//
#include <hip/hip_runtime.h>
#include <hip/hip_bf16.h>
#include <math.h>

// ---------------------------------------------------------------------------
// DFine Multi-Scale Deformable Attention for MI455X (gfx1250, wave32).
//   Kernel 1: WMMA f32 GEMM  logits = query @ [W_off | W_attn] + bias
//             (LDS-staged A tile, 18 waves/block, conflict-free LDS layout)
//   Kernel 2: softmax + bilinear gather + weighted sum (HBM-bound part)
// ---------------------------------------------------------------------------

typedef __attribute__((ext_vector_type(2))) float v2f;
typedef __attribute__((ext_vector_type(8))) float v8f;

#define DM    256            // d_model
#define NCOL  288            // 192 (off) + 96 (attn) fused columns
#define LQ    300
#define NB    32             // batch
#define SS    8400           // 80*80 + 40*40 + 20*20
#define NROW  (NB * LQ)      // 9600 query rows
#define LDA   260            // padded LDS row stride: banks = 4m+2kh+{0,1}, conflict-free

// ---------------------------------------------------------------------------
// Kernel 1: one block (18 waves, 576 threads) per 16-row stripe of query.
// A tile (16x256 f32) staged once in LDS; wave w computes N-tile w.
// WMMA f32 16x16x4 layouts:
//   A (16x4):  lane L -> row m=L&15, holds K = 2*(L>>4)+{0,1}
//   B (4x16):  lane L -> col n=L&15, holds K = 2*(L>>4)+{0,1}
//   C/D     :  VGPR r, lane L -> M = r + 8*(L>>4), N = L&15
// ---------------------------------------------------------------------------
__global__ __launch_bounds__(576)
void msda_gemm_wmma(const float* __restrict__ q,
                    const float* __restrict__ Woff,  const float* __restrict__ boff,
                    const float* __restrict__ Wattn, const float* __restrict__ battn,
                    float* __restrict__ logits)
{
    __shared__ float As[16 * LDA];            // 16.6 KB of the 320 KB WGP LDS

    const int mt   = blockIdx.x;              // 0..599  (16-row stripe)
    const int tid  = threadIdx.x;
    const int nt   = tid >> 5;                // wave id == N tile, 0..17
    const int lane = tid & 31;
    const int m    = lane & 15;               // A row == B/D column
    const int kh   = lane >> 4;               // K half selector

    // Cooperative stage of the 16x256 A tile (1024 float4 elements).
    const float* atile = q + (size_t)mt * 16 * DM;
    for (int i = tid; i < 16 * (DM / 4); i += 576) {
        const int row  = i >> 6;              // /64 float4 per row
        const int col4 = i & 63;
        *(float4*)(&As[row * LDA + col4 * 4]) = *(const float4*)(atile + row * DM + col4 * 4);
    }
    __syncthreads();

    // Column tiles 0..11 are W_off (192 cols), 12..17 are W_attn (96 cols).
    const float* Wp;  int ldw;  const float* bp;
    if (nt < 12) { Wp = Woff  + nt * 16;        ldw = 192; bp = boff  + nt * 16; }
    else         { Wp = Wattn + (nt - 12) * 16; ldw = 96;  bp = battn + (nt - 12) * 16; }

    const float* arow = &As[m * LDA];

    v8f acc = {};
#pragma unroll 8
    for (int k = 0; k < DM; k += 4) {
        v2f a = *(const v2f*)(arow + k + 2 * kh);          // ds_load_b64, no bank conflict
        v2f b;
        b.x = Wp[(size_t)(k + 2 * kh)     * ldw + m];
        b.y = Wp[(size_t)(k + 2 * kh + 1) * ldw + m];
        // (neg_a, A, neg_b, B, c_mod, C, reuse_a, reuse_b)
        acc = __builtin_amdgcn_wmma_f32_16x16x4_f32(false, a, false, b,
                                                    (short)0, acc, false, false);
    }

    const float bv   = bp[m];
    float*      orow = logits + (size_t)(mt * 16) * NCOL + nt * 16 + m;
#pragma unroll
    for (int r = 0; r < 8; ++r)
        orow[(size_t)(r + 8 * kh) * NCOL] = acc[r] + bv;
}

// ---------------------------------------------------------------------------
// Kernel 2: one block per (b,q); wave h handles head h, lane c = channel.
// Each corner gather is one coalesced 128B wave transaction
// (32 contiguous channels of one value row).
// ---------------------------------------------------------------------------
__global__ __launch_bounds__(256)
void msda_sample(const float* __restrict__ refp,
                 const float* __restrict__ vals,
                 const float* __restrict__ logits,
                 float* __restrict__ out)
{
    const int bq = blockIdx.x;            // 0..9599
    const int b  = bq / LQ;
    const int h  = threadIdx.x >> 5;      // head  (wave id)
    const int c  = threadIdx.x & 31;      // channel within head

    const float* lg = logits + (size_t)bq * NCOL;

    // Softmax over the 12 attention logits of this head (broadcast loads).
    float w[12];
    float mx = -1e30f;
#pragma unroll
    for (int p = 0; p < 12; ++p) { w[p] = lg[192 + h * 12 + p]; mx = fmaxf(mx, w[p]); }
    float s = 0.f;
#pragma unroll
    for (int p = 0; p < 12; ++p) { w[p] = __expf(w[p] - mx); s += w[p]; }
    const float inv = 1.f / s;

    const float rx = refp[(size_t)bq * 4 + 0];
    const float ry = refp[(size_t)bq * 4 + 1];
    const float rw = refp[(size_t)bq * 4 + 2];
    const float rh = refp[(size_t)bq * 4 + 3];

    // Base pointer to this (batch, head, channel) lane inside input_flatten.
    const float* vbase = vals + (size_t)b * SS * DM + h * 32 + c;

    float acc = 0.f;
#pragma unroll
    for (int p = 0; p < 12; ++p) {
        const int lvl  = p >> 2;                                   // 4 points/level
        const int Wl   = (lvl == 0) ? 80 : ((lvl == 1) ? 40 : 20);
        const int Hl   = Wl;
        const int base = (lvl == 0) ? 0  : ((lvl == 1) ? 6400 : 8000);

        const float ox = lg[h * 24 + p * 2 + 0];
        const float oy = lg[h * 24 + p * 2 + 1];
        // loc = ref_xy + off * (1/P=0.25) * ref_wh * OFFSET_SCALE(0.5)
        const float lx = rx + ox * 0.125f * rw;
        const float ly = ry + oy * 0.125f * rh;
        // grid = 2*loc-1 ; ix = ((gx+1)*W - 1)*0.5  ==  lx*W - 0.5
        const float ix = lx * (float)Wl - 0.5f;
        const float iy = ly * (float)Hl - 0.5f;

        const float fxi = floorf(ix), fyi = floorf(iy);
        const int   ix0 = (int)fxi,   iy0 = (int)fyi;
        const int   ix1 = ix0 + 1,    iy1 = iy0 + 1;
        const float fx  = ix - fxi,   fy  = iy - fyi;

        float sv = 0.f;
        if (ix0 >= 0 && ix0 < Wl && iy0 >= 0 && iy0 < Hl)
            sv += (1.f - fx) * (1.f - fy) * vbase[(size_t)(base + iy0 * Wl + ix0) * DM];
        if (ix1 >= 0 && ix1 < Wl && iy0 >= 0 && iy0 < Hl)
            sv += fx * (1.f - fy)         * vbase[(size_t)(base + iy0 * Wl + ix1) * DM];
        if (ix0 >= 0 && ix0 < Wl && iy1 >= 0 && iy1 < Hl)
            sv += (1.f - fx) * fy         * vbase[(size_t)(base + iy1 * Wl + ix0) * DM];
        if (ix1 >= 0 && ix1 < Wl && iy1 >= 0 && iy1 < Hl)
            sv += fx * fy                 * vbase[(size_t)(base + iy1 * Wl + ix1) * DM];

        acc += (w[p] * inv) * sv;
    }

    // out[b, q, h*32 + c]  (matches reference final reshape)
    out[(size_t)bq * DM + threadIdx.x] = acc;
}

// ---------------------------------------------------------------------------
extern "C" void kernel_launch(void* const* d_in, const int* in_sizes, int n_in,
                              void* d_out, int out_size, void* d_ws, size_t ws_size,
                              hipStream_t stream)
{
    const float* query = (const float*)d_in[0];
    const float* refp  = (const float*)d_in[1];
    const float* vals  = (const float*)d_in[2];
    const float* Woff  = (const float*)d_in[3];
    const float* boff  = (const float*)d_in[4];
    const float* Wattn = (const float*)d_in[5];
    const float* battn = (const float*)d_in[6];
    float*       out   = (float*)d_out;

    // logits workspace: 9600 x 288 f32 = ~11 MB, fully overwritten each call.
    float* logits = (float*)d_ws;

    // One block (18 waves) per 16-row stripe: A staged in LDS once.
    msda_gemm_wmma<<<NROW / 16, 576, 0, stream>>>(query, Woff, boff, Wattn, battn, logits);

    msda_sample<<<NROW, 256, 0, stream>>>(refp, vals, logits, out);
}